// Unfold_55791625175250
// MI455X (gfx1250) — compile-verified
//
#include <hip/hip_runtime.h>
#include <stdint.h>

// Unfold / im2col with K=3, pad=1 on (16,128,64,64) f32 -> (16,1152,4096) f32.
// Pure data movement: stage each 64x64 plane (zero-padded to 66x66) in LDS via
// CDNA5 async global->LDS DMA, then stream out 9 shifted copies with wide
// non-temporal stores. Bandwidth-bound: ~320 MB total traffic.

#define BATCH 16
#define CHAN  128
#define H     64
#define W     64
#define LDS_STRIDE 80   // floats per padded row: 66 used, padded so interior col is 16B aligned
#define LDS_ROWS   66
// LDS index of padded coordinate (py, px), py,px in [0,65]:
//   idx(py,px) = py*LDS_STRIDE + px + 3   (interior col c -> px=c+1 -> +4 : 16B aligned)

typedef float v4f __attribute__((ext_vector_type(4)));

__global__ __launch_bounds__(256) void unfold_kernel(const float* __restrict__ in,
                                                     float* __restrict__ out) {
    __shared__ __align__(16) float lds[LDS_ROWS * LDS_STRIDE];

    const int tid   = threadIdx.x;
    const int plane = blockIdx.x;                      // b*CHAN + c
    const float* src = in + (size_t)plane * (H * W);

    // ---- Zero the 1-wide halo (rows 0 & 65 full width; cols 0 & 65 rows 1..64).
    // Runs concurrently with the async DMA below (disjoint LDS addresses).
    if (tid < 66) {
        lds[0 * LDS_STRIDE + tid + 3]  = 0.0f;         // top row, px = tid
        lds[65 * LDS_STRIDE + tid + 3] = 0.0f;         // bottom row
    } else if (tid < 130) {
        int r = tid - 65;                              // 1..64
        lds[r * LDS_STRIDE + 0 + 3]  = 0.0f;           // left col  (px=0)
        lds[r * LDS_STRIDE + 65 + 3] = 0.0f;           // right col (px=65)
    }

    // ---- Async global -> LDS DMA of the 64x64 interior, 16B per op.
    // 1024 float4 tiles, 4 per thread. LDS dest is 16B aligned by construction.
#pragma unroll
    for (int i = 0; i < 4; ++i) {
        int q  = tid + 256 * i;                        // 0..1023
        int r  = q >> 4;                               // row   0..63
        int c4 = (q & 15) << 2;                        // col   0,4,...,60
        unsigned lds_off =
            (unsigned)(uintptr_t)&lds[(r + 1) * LDS_STRIDE + c4 + 4];
        uint64_t gaddr = (uint64_t)(uintptr_t)(src + r * W + c4);
        asm volatile("global_load_async_to_lds_b128 %0, %1, off"
                     :: "v"(lds_off), "v"(gaddr)
                     : "memory");
    }
    asm volatile("s_wait_asynccnt 0" ::: "memory");
    __syncthreads();

    // ---- Emit the 9 shifted copies. Each thread: 4 groups of 4 consecutive
    // output pixels; per group read an 18-float LDS neighborhood once, then do
    // 9 float4 non-temporal global stores (one per patch channel).
    float* outp = out + (size_t)plane * 9 * (H * W);
#pragma unroll
    for (int i = 0; i < 4; ++i) {
        int q  = tid + 256 * i;
        int y  = q >> 4;                               // 0..63
        int x4 = (q & 15) << 2;                        // 0..60

        float rv[3][6];
#pragma unroll
        for (int dy = 0; dy < 3; ++dy)
#pragma unroll
            for (int j = 0; j < 6; ++j)
                rv[dy][j] = lds[(y + dy) * LDS_STRIDE + x4 + 3 + j];

#pragma unroll
        for (int p = 0; p < 9; ++p) {
            const int dy = p / 3, dx = p % 3;
            v4f v = { rv[dy][dx], rv[dy][dx + 1], rv[dy][dx + 2], rv[dy][dx + 3] };
            __builtin_nontemporal_store(
                v, (v4f*)(outp + (size_t)p * (H * W) + y * W + x4));
        }
    }
}

extern "C" void kernel_launch(void* const* d_in, const int* in_sizes, int n_in,
                              void* d_out, int out_size, void* d_ws, size_t ws_size,
                              hipStream_t stream) {
    (void)in_sizes; (void)n_in; (void)out_size; (void)d_ws; (void)ws_size;
    const float* x = (const float*)d_in[0];
    float* out    = (float*)d_out;
    dim3 grid(BATCH * CHAN);   // one block per (b,c) plane = 2048 blocks
    dim3 block(256);           // 8 waves (wave32)
    unfold_kernel<<<grid, block, 0, stream>>>(x, out);
}